// VectorQuantizer_58798102282860
// MI455X (gfx1250) — compile-verified
//
#include <hip/hip_runtime.h>
#include <hip/hip_bf16.h>
#include <math.h>

typedef __attribute__((ext_vector_type(2))) float v2f;
typedef __attribute__((ext_vector_type(8))) float v8f;

#define N_ROWS  65536
#define D_DIM   64
#define K_CODES 512

// d_out layout (flat, reference tuple return order, all f32):
static constexpr size_t Q_OFF    = 0;                       // quantized: 65536*64
static constexpr size_t LOSS_OFF = 4194304;                 // loss scalar
static constexpr size_t PERP_OFF = 4194305;                 // perplexity scalar
static constexpr size_t ENC_OFF  = 4194306;                 // encodings: 65536*512
static constexpr size_t IDX_OFF  = 37748738;                // encoding_indices: 65536
static constexpr size_t DIST_OFF = 37814274;                // distances: 65536*512

// ---------------------------------------------------------------------------
// ||e_k||^2 for each of the 512 codes
// ---------------------------------------------------------------------------
__global__ __launch_bounds__(256) void vq_esq_kernel(const float* __restrict__ E,
                                                     float* __restrict__ esq) {
    int k = blockIdx.x * blockDim.x + threadIdx.x;
    if (k < K_CODES) {
        float s = 0.0f;
        #pragma unroll 8
        for (int d = 0; d < D_DIM; ++d) {
            float e = E[(size_t)d * K_CODES + k];
            s += e * e;
        }
        esq[k] = s;
    }
}

// ---------------------------------------------------------------------------
// Fused: distances (WMMA f32 16x16x4), argmin, one-hot, gather, SSE, histogram
// One block = 16 rows; 8 waves each own 4 column tiles of 16 (K=512 total).
// ---------------------------------------------------------------------------
__global__ __launch_bounds__(256) void vq_main_kernel(const float* __restrict__ X,
                                                      const float* __restrict__ E,
                                                      const float* __restrict__ esq,
                                                      float* __restrict__ counts,
                                                      float* __restrict__ sse,
                                                      float* __restrict__ out) {
    __shared__ float sXsq[16];
    __shared__ float sMin[8][16];
    __shared__ int   sIdx[8][16];
    __shared__ int   sFinal[16];
    __shared__ float sWred[8];

    const int tid      = threadIdx.x;
    const int wave     = tid >> 5;
    const int lane     = tid & 31;
    const int row_base = blockIdx.x * 16;

    // ||x_row||^2 for the 16 rows of this block
    if (tid < 16) {
        const float* xr = X + (size_t)(row_base + tid) * D_DIM;
        float s = 0.0f;
        #pragma unroll 8
        for (int d = 0; d < D_DIM; ++d) s += xr[d] * xr[d];
        sXsq[tid] = s;
    }
    __syncthreads();

    const int m     = lane & 15;   // row (A) / col (B,C) within tile
    const int khalf = lane >> 4;   // which K-pair of the 16x4 fragment

    // Preload A fragments once per wave: 16 steps of K=4 cover K=64.
    v2f af[16];
    {
        const float* xr = X + (size_t)(row_base + m) * D_DIM;
        #pragma unroll
        for (int kk = 0; kk < 16; ++kk) {
            const int kb = kk * 4 + khalf * 2;
            af[kk].x = xr[kb];
            af[kk].y = xr[kb + 1];
        }
    }

    float bm[8];
    int   bi[8];
    #pragma unroll
    for (int i = 0; i < 8; ++i) { bm[i] = __builtin_inff(); bi[i] = 0x7fffffff; }

    float* dist_out = out + DIST_OFF;

    for (int t = 0; t < 4; ++t) {
        const int col_base = (wave * 4 + t) * 16;
        const int n        = col_base + m;

        // B fragment: 4x16 K-major slice of E (E is [D=64][K=512] row-major)
        v2f bf[16];
        #pragma unroll
        for (int kk = 0; kk < 16; ++kk) {
            const int kb = kk * 4 + khalf * 2;
            bf[kk].x = E[(size_t)kb * K_CODES + n];
            bf[kk].y = E[(size_t)(kb + 1) * K_CODES + n];
        }

        v8f c = {};
        #pragma unroll
        for (int kk = 0; kk < 16; ++kk) {
            c = __builtin_amdgcn_wmma_f32_16x16x4_f32(
                    false, af[kk], false, bf[kk], (short)0, c, false, false);
        }

        const float esq_n = esq[n];
        #pragma unroll
        for (int i = 0; i < 8; ++i) {
            const int  mr   = i + 8 * khalf;           // C/D row for this lane
            const float dst = sXsq[mr] - 2.0f * c[i] + esq_n;
            dist_out[(size_t)(row_base + mr) * K_CODES + n] = dst;
            if (dst < bm[i] || (dst == bm[i] && n < bi[i])) { bm[i] = dst; bi[i] = n; }
        }
    }

    // Reduce (min,idx) across the 16 lanes of each half-wave (wave32)
    #pragma unroll
    for (int i = 0; i < 8; ++i) {
        float v  = bm[i];
        int   ix = bi[i];
        #pragma unroll
        for (int msk = 1; msk <= 8; msk <<= 1) {
            float ov = __shfl_xor(v, msk, 32);
            int   oi = __shfl_xor(ix, msk, 32);
            if (ov < v || (ov == v && oi < ix)) { v = ov; ix = oi; }
        }
        bm[i] = v; bi[i] = ix;
    }
    if ((lane & 15) == 0) {
        #pragma unroll
        for (int i = 0; i < 8; ++i) {
            sMin[wave][i + 8 * khalf] = bm[i];
            sIdx[wave][i + 8 * khalf] = bi[i];
        }
    }
    __syncthreads();

    // Cross-wave argmin per row; write indices + histogram
    if (tid < 16) {
        float v  = sMin[0][tid];
        int   ix = sIdx[0][tid];
        #pragma unroll
        for (int w = 1; w < 8; ++w) {
            float ov = sMin[w][tid];
            int   oi = sIdx[w][tid];
            if (ov < v || (ov == v && oi < ix)) { v = ov; ix = oi; }
        }
        sFinal[tid] = ix;
        out[IDX_OFF + row_base + tid] = (float)ix;
        atomicAdd(&counts[ix], 1.0f);
    }
    __syncthreads();

    // Quantized gather (straight-through output == codebook row) + local SSE
    float local = 0.0f;
    {
        int e = tid * 4;                 // 16 rows * 64 dims / 256 threads
        #pragma unroll
        for (int j = 0; j < 4; ++j, ++e) {
            const int r  = e >> 6;
            const int d  = e & 63;
            const int ix = sFinal[r];
            const float q  = E[(size_t)d * K_CODES + ix];
            const float xv = X[(size_t)(row_base + r) * D_DIM + d];
            out[Q_OFF + (size_t)(row_base + r) * D_DIM + d] = q;
            const float df = q - xv;
            local += df * df;
        }
    }
    #pragma unroll
    for (int msk = 1; msk < 32; msk <<= 1) local += __shfl_xor(local, msk, 32);
    if (lane == 0) sWred[wave] = local;
    __syncthreads();
    if (tid == 0) {
        float s = 0.0f;
        #pragma unroll
        for (int w = 0; w < 8; ++w) s += sWred[w];
        atomicAdd(sse, s);
    }

    // One-hot encodings for the 16 rows (16*512 floats)
    float* enc = out + ENC_OFF + (size_t)row_base * K_CODES;
    for (int e = tid; e < 16 * K_CODES; e += 256) {
        const int r   = e >> 9;
        const int col = e & 511;
        enc[e] = (col == sFinal[r]) ? 1.0f : 0.0f;
    }
}

// ---------------------------------------------------------------------------
// loss + perplexity scalars
// ---------------------------------------------------------------------------
__global__ __launch_bounds__(512) void vq_finalize_kernel(const float* __restrict__ counts,
                                                          const float* __restrict__ sse,
                                                          float* __restrict__ out) {
    __shared__ float sred[16];
    const int tid  = threadIdx.x;
    const int wave = tid >> 5;
    const int lane = tid & 31;

    const float p    = counts[tid] * (1.0f / (float)N_ROWS);
    float       term = p * logf(p + 1e-10f);
    #pragma unroll
    for (int msk = 1; msk < 32; msk <<= 1) term += __shfl_xor(term, msk, 32);
    if (lane == 0) sred[wave] = term;
    __syncthreads();
    if (tid == 0) {
        float s = 0.0f;
        #pragma unroll
        for (int w = 0; w < 16; ++w) s += sred[w];
        out[PERP_OFF] = expf(-s);
        // q_latent_loss == e_latent_loss numerically -> (1 + 0.25) * mean sq err
        out[LOSS_OFF] = sse[0] * 1.25f / (float)((size_t)N_ROWS * D_DIM);
    }
}

// ---------------------------------------------------------------------------
extern "C" void kernel_launch(void* const* d_in, const int* in_sizes, int n_in,
                              void* d_out, int out_size, void* d_ws, size_t ws_size,
                              hipStream_t stream) {
    const float* X = (const float*)d_in[0];   // (64,32,32,64) f32
    // d_in[1] = context, unused
    const float* E = (const float*)d_in[2];   // (64,512) f32
    float* out = (float*)d_out;

    float* esq    = (float*)d_ws;             // [512]
    float* counts = esq + K_CODES;            // [512]
    float* sse    = counts + K_CODES;         // [1]

    // zero histogram + sse (graph-capturable memset node)
    hipMemsetAsync(counts, 0, (K_CODES + 1) * sizeof(float), stream);

    vq_esq_kernel<<<2, 256, 0, stream>>>(E, esq);
    vq_main_kernel<<<N_ROWS / 16, 256, 0, stream>>>(X, E, esq, counts, sse, out);
    vq_finalize_kernel<<<1, 512, 0, stream>>>(counts, sse, out);
}